// OceanNetworkGAT_90391881712254
// MI455X (gfx1250) — compile-verified
//
#include <hip/hip_runtime.h>
#include <hip/hip_bf16.h>
#include <math.h>

// ---------------------------------------------------------------------------
// GAT forward on gfx1250 (MI455X, wave32).
//   layer: Wh = h@W (fp32 WMMA 16x16x4), s_src/s_dst per node,
//          global max over edges, single fused edge scatter pass
//          (unnormalized accumulate), per-node normalize + ELU.
//   head : per-node 32x32 gelu MLP + 32->1 projection, wave-per-node.
// ---------------------------------------------------------------------------

#define N_NODES 50000
#define N_EDGES 800000
#define NPAD    50048          // multiple of 64 so WMMA grids need no masking

typedef __attribute__((ext_vector_type(2))) float v2f;
typedef __attribute__((ext_vector_type(8))) float v8f;

// -------- float <-> order-preserving uint (for atomicMax on floats) --------
__device__ __forceinline__ unsigned f2ord(float f) {
  unsigned u = __float_as_uint(f);
  return (u & 0x80000000u) ? ~u : (u | 0x80000000u);
}
__device__ __forceinline__ float ord2f(unsigned o) {
  unsigned u = (o & 0x80000000u) ? (o & 0x7FFFFFFFu) : ~o;
  return __uint_as_float(u);
}

__device__ __forceinline__ float lrelu02(float x) { return x > 0.0f ? x : 0.2f * x; }

__device__ __forceinline__ float wave_sum(float v) {
  for (int o = 16; o > 0; o >>= 1) v += __shfl_xor(v, o, 32);
  return v;
}
__device__ __forceinline__ float wave_max(float v) {
  for (int o = 16; o > 0; o >>= 1) v = fmaxf(v, __shfl_xor(v, o, 32));
  return v;
}

// ------------------------------- zero fill ---------------------------------
__global__ void zero_ws_kernel(float* __restrict__ p, size_t n) {
  size_t i = (size_t)blockIdx.x * blockDim.x + threadIdx.x;
  size_t stride = (size_t)gridDim.x * blockDim.x;
  for (; i < n; i += stride) p[i] = 0.0f;   // bits==0 also == ordered(-NaN) for max slots
}

// ---------------- layer 1 GEMM: x[50000x8] @ W1[8x128] ---------------------
// grid = 3125 blocks x 256 threads (8 waves). Wave w -> 16x16 tile, cols w*16.
__global__ void gemm1_wmma(const float* __restrict__ x, const float* __restrict__ W,
                           float* __restrict__ Wh) {
  const int tid = threadIdx.x, wv = tid >> 5, lane = tid & 31;
  const int row0 = blockIdx.x * 16;
  const int col0 = wv * 16;
  const int m = lane & 15, lh = lane >> 4;   // lane half selects K pair (ISA A 16x4 layout)
  const int row = row0 + m;
  v8f c = {};
  for (int kb = 0; kb < 8; kb += 4) {
    const int k = kb + lh * 2;
    v2f a; a.x = x[row * 8 + k];            a.y = x[row * 8 + k + 1];
    v2f b; b.x = W[k * 128 + col0 + m];     b.y = W[(k + 1) * 128 + col0 + m];
    c = __builtin_amdgcn_wmma_f32_16x16x4_f32(false, a, false, b, (short)0, c, false, false);
  }
  // C/D layout: VGPR j -> M = lh*8 + j, N = lane&15
  for (int j = 0; j < 8; ++j)
    Wh[(row0 + lh * 8 + j) * 128 + col0 + m] = c[j];
}

// ---------------- layer 2 GEMM: h[NPAD x128] @ W2[128x32] ------------------
// grid = 782 blocks x 256 threads. Wave -> (mtile 0..3, ntile 0..1).
__global__ void gemm2_wmma(const float* __restrict__ h, const float* __restrict__ W,
                           float* __restrict__ Wh) {
  const int tid = threadIdx.x, wv = tid >> 5, lane = tid & 31;
  const int row0 = blockIdx.x * 64 + (wv >> 1) * 16;
  const int col0 = (wv & 1) * 16;
  const int m = lane & 15, lh = lane >> 4;
  const int row = row0 + m;
  v8f c = {};
  for (int kb = 0; kb < 128; kb += 4) {
    const int k = kb + lh * 2;
    v2f a; a.x = h[row * 128 + k];          a.y = h[row * 128 + k + 1];
    v2f b; b.x = W[k * 32 + col0 + m];      b.y = W[(k + 1) * 32 + col0 + m];
    c = __builtin_amdgcn_wmma_f32_16x16x4_f32(false, a, false, b, (short)0, c, false, false);
  }
  for (int j = 0; j < 8; ++j)
    Wh[(row0 + lh * 8 + j) * 32 + col0 + m] = c[j];
}

// --------- per-node attention scalars: s_src = Wh.a[:D], s_dst = Wh.a[D:] --
__global__ void node_dots(const float* __restrict__ Wh, const float* __restrict__ a,
                          int D, int n, float* __restrict__ ss, float* __restrict__ sd) {
  const int lane = threadIdx.x & 31;
  const int node = (blockIdx.x * blockDim.x + threadIdx.x) >> 5;
  if (node >= n) return;
  float as = 0.0f, ad = 0.0f;
  for (int k = lane; k < D; k += 32) {
    const float w = Wh[(size_t)node * D + k];
    as += w * a[k];
    ad += w * a[D + k];
  }
  as = wave_sum(as);
  ad = wave_sum(ad);
  if (lane == 0) { ss[node] = as; sd[node] = ad; }
}

// ------------- global max of leaky_relu(s_src[u] + s_dst[v]) ---------------
__global__ void edge_max(const float* __restrict__ ss, const float* __restrict__ sd,
                         const int* __restrict__ src, const int* __restrict__ dst,
                         int E, unsigned* __restrict__ mslot) {
  const int e = blockIdx.x * blockDim.x + threadIdx.x;
  float v = -INFINITY;
  if (e < E) v = lrelu02(ss[src[e]] + sd[dst[e]]);
  v = wave_max(v);
  if ((threadIdx.x & 31) == 0) atomicMax(mslot, f2ord(v));
}

// -------- fused edge pass: out[v] += e*Wh[u];  sum[v] += e  (wave/edge) ----
__global__ void edge_agg(const float* __restrict__ Wh, const float* __restrict__ ss,
                         const float* __restrict__ sd, const int* __restrict__ src,
                         const int* __restrict__ dst, int E, int D,
                         const unsigned* __restrict__ mslot,
                         float* __restrict__ out, float* __restrict__ sum) {
  const int lane = threadIdx.x & 31;
  const int e = (blockIdx.x * blockDim.x + threadIdx.x) >> 5;
  if (e >= E) return;
  const int u = src[e], v = dst[e];
  const float mx = ord2f(*mslot);
  const float w = __expf(lrelu02(ss[u] + sd[v]) - mx);
  if (lane == 0) atomicAdd(&sum[v], w);
  for (int k = lane; k < D; k += 32)
    atomicAdd(&out[(size_t)v * D + k], w * Wh[(size_t)u * D + k]);
}

// ---------------- in-place normalize + ELU: h = elu(out/(sum+eps)) ---------
__global__ void elu_norm(float* __restrict__ out, const float* __restrict__ sum,
                         int n, int D) {
  const size_t idx = (size_t)blockIdx.x * blockDim.x + threadIdx.x;
  if (idx >= (size_t)n * D) return;
  const int node = (int)(idx / D);
  const float v = out[idx] / (sum[node] + 1e-9f);
  out[idx] = v > 0.0f ? v : (expf(v) - 1.0f);
}

// --------------- MLP head: gelu(h@hw1+hb1)@hw2+hb2, wave per node ----------
__global__ void head_kernel(const float* __restrict__ h, const float* __restrict__ hw1,
                            const float* __restrict__ hb1, const float* __restrict__ hw2,
                            const float* __restrict__ hb2, float* __restrict__ scores,
                            int n) {
  const int lane = threadIdx.x & 31;
  const int node = (blockIdx.x * blockDim.x + threadIdx.x) >> 5;
  if (node >= n) return;
  const float mine = h[(size_t)node * 32 + lane];
  float acc = hb1[lane];
  for (int k = 0; k < 32; ++k)
    acc += __shfl(mine, k, 32) * hw1[k * 32 + lane];
  // exact gelu: 0.5*x*(1+erf(x/sqrt(2)))
  const float z = 0.5f * acc * (1.0f + erff(acc * 0.70710678118654752f));
  float s = wave_sum(z * hw2[lane]);
  if (lane == 0) scores[node] = s + hb2[0];
}

// ---------------------------------------------------------------------------
extern "C" void kernel_launch(void* const* d_in, const int* in_sizes, int n_in,
                              void* d_out, int out_size, void* d_ws, size_t ws_size,
                              hipStream_t stream) {
  const float* x   = (const float*)d_in[0];
  const int*   ei  = (const int*)  d_in[1];
  const float* W1  = (const float*)d_in[2];
  const float* a1  = (const float*)d_in[3];
  const float* W2  = (const float*)d_in[4];
  const float* a2  = (const float*)d_in[5];
  const float* hw1 = (const float*)d_in[6];
  const float* hb1 = (const float*)d_in[7];
  const float* hw2 = (const float*)d_in[8];
  const float* hb2 = (const float*)d_in[9];
  float* scores = (float*)d_out;

  const int N = N_NODES;
  const int E = in_sizes[1] / 2;
  const int* src = ei;
  const int* dst = ei + E;

  // -------- workspace carve-up (all chunks 16B aligned) --------
  float* p    = (float*)d_ws;
  float* Wh1  = p;  p += (size_t)NPAD * 128;   // layer-1 transformed features
  float* out1 = p;  p += (size_t)NPAD * 128;   // layer-1 aggregate -> h1 (in place)
  float* Wh2  = p;  p += (size_t)NPAD * 32;    // layer-2 transformed features
  float* out2 = p;  p += (size_t)NPAD * 32;    // layer-2 aggregate -> h2 (in place)
  float* s1s  = p;  p += NPAD;
  float* s1d  = p;  p += NPAD;
  float* sum1 = p;  p += NPAD;
  float* s2s  = p;  p += NPAD;
  float* s2d  = p;  p += NPAD;
  float* sum2 = p;  p += NPAD;
  unsigned* mslots = (unsigned*)p;             // [0]=layer1 max, [1]=layer2 max
  const size_t totalFloats = (size_t)(p - (float*)d_ws) + 2;

  // -------- init (harness does not re-poison between replays) --------
  zero_ws_kernel<<<4096, 256, 0, stream>>>((float*)d_ws, totalFloats);

  // -------- layer 1 (D = 128) --------
  gemm1_wmma<<<N / 16, 256, 0, stream>>>(x, W1, Wh1);                       // 3125 blocks
  node_dots<<<(N * 32 + 255) / 256, 256, 0, stream>>>(Wh1, a1, 128, N, s1s, s1d);
  edge_max<<<(E + 255) / 256, 256, 0, stream>>>(s1s, s1d, src, dst, E, &mslots[0]);
  edge_agg<<<(int)(((size_t)E * 32 + 255) / 256), 256, 0, stream>>>(
      Wh1, s1s, s1d, src, dst, E, 128, &mslots[0], out1, sum1);
  elu_norm<<<(N * 128 + 255) / 256, 256, 0, stream>>>(out1, sum1, N, 128);

  // -------- layer 2 (D = 32), reads padded (zeroed) rows of out1 --------
  gemm2_wmma<<<NPAD / 64, 256, 0, stream>>>(out1, W2, Wh2);                 // 782 blocks
  node_dots<<<(N * 32 + 255) / 256, 256, 0, stream>>>(Wh2, a2, 32, N, s2s, s2d);
  edge_max<<<(E + 255) / 256, 256, 0, stream>>>(s2s, s2d, src, dst, E, &mslots[1]);
  edge_agg<<<(int)(((size_t)E * 32 + 255) / 256), 256, 0, stream>>>(
      Wh2, s2s, s2d, src, dst, E, 32, &mslots[1], out2, sum2);
  elu_norm<<<(N * 32 + 255) / 256, 256, 0, stream>>>(out2, sum2, N, 32);

  // -------- MLP head --------
  head_kernel<<<(N * 32 + 255) / 256, 256, 0, stream>>>(out2, hw1, hb1, hw2, hb2,
                                                        scores, N);
}